// QuantumHybridHead_65481071407668
// MI455X (gfx1250) — compile-verified
//
#include <hip/hip_runtime.h>
#include <hip/hip_bf16.h>

// ---------------------------------------------------------------------------
// QuantumHybridHead on gfx1250 (MI455X).
//
// Math: out[b,w] = sum_j sign(w,j) * | (V psi_b)_j |^2
//   - V: fixed 16x16 complex unitary from theta (built once per launch).
//   - psi_b: product state from per-sample encoder; rz(w2) is a global phase
//     (exactly cancels in |.|^2), leaving only 8 live columns of V.
//   - Phi = Vc(16x8) x Psi'(8 x 16samples) done as f32 WMMA 16x16x4
//     (complex = 4 real matmuls: Re = Vr*Pr + (-Vi)*Pi ; Im = Vr*Pi + Vi*Pr).
// ---------------------------------------------------------------------------

typedef __attribute__((ext_vector_type(2))) float v2f;
typedef __attribute__((ext_vector_type(4))) float v4f;
typedef __attribute__((ext_vector_type(8))) float v8f;

// ---------------------------------------------------------------------------
// Setup: one wave builds the compacted trainable-circuit matrix Vc (16x8).
// Lane j' (0..7) evolves basis vector e_{jmap(j')} through the fixed gate
// sequence; the result is column j' of Vc. Stored row-major in ws:
//   ws[0..127]   = Re(Vc),  ws[128..255] = Im(Vc),  ws[256..383] = -Im(Vc)
// ---------------------------------------------------------------------------
__global__ void qhh_setup(const float* __restrict__ theta, float* __restrict__ ws) {
  const int jp = threadIdx.x;
  if (jp >= 8) return;
  // j' bits (q0,q1,q3) -> full basis index q0*8 + q1*4 + q3 (q2 = 0)
  const int j0 = ((jp & 4) << 1) | ((jp & 2) << 1) | (jp & 1);

  float sr[16], si[16];
#pragma unroll
  for (int i = 0; i < 16; ++i) { sr[i] = (i == j0) ? 1.f : 0.f; si[i] = 0.f; }

  auto gate = [&](int mask, float u00r, float u00i, float u01r, float u01i,
                  float u10r, float u10i, float u11r, float u11i) {
#pragma unroll
    for (int j = 0; j < 16; ++j) {
      if (j & mask) continue;
      const int j1 = j | mask;
      float ar = sr[j], ai = si[j], br = sr[j1], bi = si[j1];
      sr[j]  = u00r * ar - u00i * ai + u01r * br - u01i * bi;
      si[j]  = u00r * ai + u00i * ar + u01r * bi + u01i * br;
      sr[j1] = u10r * ar - u10i * ai + u11r * br - u11i * bi;
      si[j1] = u10r * ai + u10i * ar + u11r * bi + u11i * br;
    }
  };
  auto cnot = [&](int mc, int mt) {
#pragma unroll
    for (int j = 0; j < 16; ++j) {
      if ((j & mc) && !(j & mt)) {
        const int j1 = j | mt;
        float tr = sr[j], ti = si[j];
        sr[j] = sr[j1]; si[j] = si[j1];
        sr[j1] = tr;    si[j1] = ti;
      }
    }
  };

  int k = 0;
#pragma unroll
  for (int q = 0; q < 4; ++q) {
    const int mask = 8 >> q;
    float c, s;
    __sincosf(0.5f * theta[k], &s, &c); ++k;             // rx
    gate(mask, c, 0.f, 0.f, -s, 0.f, -s, c, 0.f);
    __sincosf(0.5f * theta[k], &s, &c); ++k;             // ry
    gate(mask, c, 0.f, -s, 0.f, s, 0.f, c, 0.f);
    __sincosf(0.5f * theta[k], &s, &c); ++k;             // rz
    gate(mask, c, -s, 0.f, 0.f, 0.f, 0.f, c, s);
  }
  cnot(8 >> 0, 8 >> 1);
  cnot(8 >> 1, 8 >> 2);
  cnot(8 >> 2, 8 >> 3);
  cnot(8 >> 3, 8 >> 0);
#pragma unroll
  for (int q = 0; q < 4; ++q) {
    const int mask = 8 >> q;
    float c, s;
    __sincosf(0.5f * theta[12 + q], &s, &c);             // rx
    gate(mask, c, 0.f, 0.f, -s, 0.f, -s, c, 0.f);
  }

#pragma unroll
  for (int i = 0; i < 16; ++i) {
    ws[i * 8 + jp]       = sr[i];
    ws[128 + i * 8 + jp] = si[i];
    ws[256 + i * 8 + jp] = -si[i];
  }
}

// ---------------------------------------------------------------------------
// Main: each wave processes 32 samples (two 16-sample WMMA N-tiles).
// A operand (16x8, K split as 2 chunks of 4): lane(n,h) VGPR p of chunk c
// holds Vc[n][4c + 2h + p].  B operand: lane(n,h) holds psi' rows
// K' = 4c + 2h + p for sample n of the tile; with K' bits (q0,q1,q3)=(c,h,p)
// the entries are closed-form in the encoder trig values.
// D layout: VGPR r, lane(n,h) = Phi[basis r+8h][sample n].
// ---------------------------------------------------------------------------
__global__ void __launch_bounds__(256) qhh_main(const float4* __restrict__ in4,
                                                const float* __restrict__ ws,
                                                float* __restrict__ out,
                                                int nsamp, int nchunks) {
  const int lane = threadIdx.x & 31;
  const int n = lane & 15;
  const int h = lane >> 4;
  const int waveId = (blockIdx.x * blockDim.x + threadIdx.x) >> 5;
  const int nWaves = (gridDim.x * blockDim.x) >> 5;

  // Load A-layout registers for Vr, Vi, -Vi (2 K-chunks each).
  v2f Ar[2], Ai[2], An[2];
#pragma unroll
  for (int c = 0; c < 2; ++c) {
    const float* p = ws + n * 8 + 4 * c + 2 * h;
    Ar[c] = v2f{p[0],   p[1]};
    Ai[c] = v2f{p[128], p[129]};
    An[c] = v2f{p[256], p[257]};
  }

  for (int chunk = waveId; chunk < nchunks; chunk += nWaves) {
    const int base = chunk * 32;
    v4f res[2];

#pragma unroll
    for (int t = 0; t < 2; ++t) {
      const int samp = base + t * 16 + n;
      const int sampc = (samp < nsamp) ? samp : (nsamp - 1);
      const float4 wv = in4[sampc];

      float sa, ca, sb, cb, sd, cd;
      __sincosf(0.5f * wv.x, &sa, &ca);   // rx(w0) on qubit 0
      __sincosf(0.5f * wv.y, &sb, &cb);   // ry(w1) on qubit 1
      __sincosf(0.5f * wv.w, &sd, &cd);   // rx(w3) on qubit 3  (rz(w2): global phase, drops)

      const float bq = h ? sb : cb;       // q1 = h in this lane's K rows
      const float t0 = bq * ca, t1 = bq * sa;
      v2f Br[2], Bi[2];
      Br[0] = v2f{ t0 * cd,  0.f };  Bi[0] = v2f{ 0.f, -t0 * sd };
      Br[1] = v2f{ 0.f, -t1 * sd };  Bi[1] = v2f{ -t1 * cd, 0.f };

      v8f accR = {0.f, 0.f, 0.f, 0.f, 0.f, 0.f, 0.f, 0.f};
      v8f accI = {0.f, 0.f, 0.f, 0.f, 0.f, 0.f, 0.f, 0.f};
#pragma unroll
      for (int c = 0; c < 2; ++c) {
        // Re(Phi) = Vr*Pr + (-Vi)*Pi
        accR = __builtin_amdgcn_wmma_f32_16x16x4_f32(false, Ar[c], false, Br[c],
                                                     (short)0, accR, false, false);
        accR = __builtin_amdgcn_wmma_f32_16x16x4_f32(false, An[c], false, Bi[c],
                                                     (short)0, accR, false, false);
        // Im(Phi) = Vr*Pi + Vi*Pr
        accI = __builtin_amdgcn_wmma_f32_16x16x4_f32(false, Ar[c], false, Bi[c],
                                                     (short)0, accI, false, false);
        accI = __builtin_amdgcn_wmma_f32_16x16x4_f32(false, Ai[c], false, Br[c],
                                                     (short)0, accI, false, false);
      }

      // probs for basis rows m = r + 8h of this lane's sample
      float pr[8];
#pragma unroll
      for (int r = 0; r < 8; ++r) pr[r] = accR[r] * accR[r] + accI[r] * accI[r];

      // Z-sign reduction: m bits (wire0=MSB): bit3=h, bit2=r>>2, bit1=(r>>1)&1, bit0=r&1
      const float e01 = pr[0] + pr[1], e23 = pr[2] + pr[3];
      const float e45 = pr[4] + pr[5], e67 = pr[6] + pr[7];
      const float q0123 = e01 + e23, q4567 = e45 + e67;
      float w0 = q0123 + q4567;
      float w1 = q0123 - q4567;
      float w2 = (e01 - e23) + (e45 - e67);
      float w3 = (pr[0] - pr[1]) + (pr[2] - pr[3]) + (pr[4] - pr[5]) + (pr[6] - pr[7]);
      w0 = h ? -w0 : w0;                  // wire0 sign depends on lane half

      // combine the two basis halves (lanes n and n+16 hold the same sample)
      w0 += __shfl_xor(w0, 16, 32);
      w1 += __shfl_xor(w1, 16, 32);
      w2 += __shfl_xor(w2, 16, 32);
      w3 += __shfl_xor(w3, 16, 32);
      res[t] = v4f{w0, w1, w2, w3};
    }

    // Full-lane store: half h stores tile h's sample.
    const int mySamp = base + h * 16 + n;
    const v4f r = h ? res[1] : res[0];
    if (mySamp < nsamp) {
      *reinterpret_cast<v4f*>(out + mySamp * 4) = r;
    }
  }
}

extern "C" void kernel_launch(void* const* d_in, const int* in_sizes, int n_in,
                              void* d_out, int out_size, void* d_ws, size_t ws_size,
                              hipStream_t stream) {
  const float* inputs = (const float*)d_in[0];  // [B,4] f32
  const float* theta  = (const float*)d_in[1];  // [16]  f32
  float* out = (float*)d_out;                   // [B,4] f32
  float* ws  = (float*)d_ws;                    // 384 floats used

  const int nsamp = in_sizes[0] / 4;
  const int nchunks = (nsamp + 31) / 32;

  qhh_setup<<<1, 32, 0, stream>>>(theta, ws);

  int blocks = (nchunks + 7) / 8;               // 8 waves / block, 1 chunk / wave
  if (blocks > 2048) blocks = 2048;             // cap; grid-stride covers the rest
  if (blocks < 1) blocks = 1;
  qhh_main<<<blocks, 256, 0, stream>>>((const float4*)inputs, ws, out, nsamp, nchunks);
}